// Bottleneck_33809982554703
// MI455X (gfx1250) — compile-verified
//
#include <hip/hip_runtime.h>

typedef float v2f __attribute__((ext_vector_type(2)));
typedef float v8f __attribute__((ext_vector_type(8)));

constexpr int Bq     = 8;
constexpr int Jq     = 3;
constexpr int Nq     = 2048;
constexpr int Cq     = 32;
constexpr int Fq     = 32;
constexpr int INTERq = 128;
constexpr int JCq    = Jq * Cq;   // 96

// ---------------- Kernel 1: gmul GEMM (fp32 WMMA) + fused W pass-through copy ---------
// Per block: 4 waves, 64 rows (n) x 32 cols (c), K loop over m in chunks of 32.
constexpr int KCHUNK = 32;
constexpr int ROWS   = 64;
constexpr int APAD   = 36;   // LDS row stride in floats: 144B => 16B aligned, conflict-free

__global__ __launch_bounds__(128) void gmul_wmma_kernel(
    const float* __restrict__ Wg, const float* __restrict__ xg,
    float* __restrict__ Wout, float* __restrict__ out1)
{
    __shared__ float As[ROWS * APAD];

    const int tid    = threadIdx.x;
    const int wv     = tid >> 5;
    const int lane   = tid & 31;
    const int laneHi = lane >> 4;   // 0 or 1
    const int laneLo = lane & 15;

    const int rb   = blockIdx.x & 31;   // 32 row-blocks of 64 rows
    const int bj   = blockIdx.x >> 5;   // 0..23
    const int b    = bj / Jq;
    const int j    = bj % Jq;
    const int row0 = rb * ROWS;

    const float* Wbase = Wg   + ((size_t)(b * Jq + j) * Nq + row0) * Nq;
    float*       Obase = Wout + ((size_t)(b * Jq + j) * Nq + row0) * Nq;
    const float* xb    = xg   + (size_t)b * Cq * Nq;

    v8f acc0 = {};   // cols 0..15
    v8f acc1 = {};   // cols 16..31

    for (int k0 = 0; k0 < Nq; k0 += KCHUNK) {
        // ---- stage A (64x32) into LDS; fuse the W -> d_out copy on the same data ----
        #pragma unroll
        for (int i = 0; i < 4; ++i) {
            const int idx = tid + i * 128;          // 0..511 float4 tiles
            const int row = idx >> 3;               // 64 rows
            const int kq  = (idx & 7) << 2;         // 8 float4 per row
            const float4 v = *reinterpret_cast<const float4*>(
                Wbase + (size_t)row * Nq + k0 + kq);
            *reinterpret_cast<float4*>(&As[row * APAD + kq]) = v;
            *reinterpret_cast<float4*>(Obase + (size_t)row * Nq + k0 + kq) = v;
        }
        __syncthreads();

        const int rowW = wv * 16;
        #pragma unroll
        for (int kk = 0; kk < KCHUNK / 4; ++kk) {
            // A 16x4 fp32 fragment: lanes 0-15 -> K=0,1 ; lanes 16-31 -> K=2,3
            const int kOff = kk * 4 + laneHi * 2;
            const v2f a = *reinterpret_cast<const v2f*>(
                &As[(rowW + laneLo) * APAD + kOff]);
            // B 4x16 fp32 fragments straight from x (L2-resident)
            const int m = k0 + kOff;
            const v2f b0 = *reinterpret_cast<const v2f*>(
                xb + (size_t)laneLo * Nq + m);
            const v2f b1 = *reinterpret_cast<const v2f*>(
                xb + (size_t)(16 + laneLo) * Nq + m);
            acc0 = __builtin_amdgcn_wmma_f32_16x16x4_f32(
                false, a, false, b0, (short)0, acc0, false, false);
            acc1 = __builtin_amdgcn_wmma_f32_16x16x4_f32(
                false, a, false, b1, (short)0, acc1, false, false);
        }
        __syncthreads();
    }

    // ---- epilogue: D layout => lane = col, VGPR r = row (+8 for hi lanes) ----
    // acc element r is n = n0 + r  => rows contiguous: two float4 stores per fragment
    const int n0 = row0 + wv * 16 + laneHi * 8;
    {
        union { v8f v; float f[8]; } u; u.v = acc0;
        float* p = out1 + ((size_t)b * JCq + j * Cq + laneLo) * Nq + n0;
        float4 lo; lo.x = u.f[0]; lo.y = u.f[1]; lo.z = u.f[2]; lo.w = u.f[3];
        float4 hi; hi.x = u.f[4]; hi.y = u.f[5]; hi.z = u.f[6]; hi.w = u.f[7];
        *reinterpret_cast<float4*>(p)     = lo;
        *reinterpret_cast<float4*>(p + 4) = hi;
    }
    {
        union { v8f v; float f[8]; } u; u.v = acc1;
        float* p = out1 + ((size_t)b * JCq + j * Cq + 16 + laneLo) * Nq + n0;
        float4 lo; lo.x = u.f[0]; lo.y = u.f[1]; lo.z = u.f[2]; lo.w = u.f[3];
        float4 hi; hi.x = u.f[4]; hi.y = u.f[5]; hi.z = u.f[6]; hi.w = u.f[7];
        *reinterpret_cast<float4*>(p)     = lo;
        *reinterpret_cast<float4*>(p + 4) = hi;
    }
}

// ---------------- Kernel 2/4: BN statistics -> per-channel scale/shift ----------------
__global__ __launch_bounds__(256) void bn_stats_kernel(
    const float* __restrict__ data, const float* __restrict__ gamma,
    const float* __restrict__ beta, int CH,
    float* __restrict__ scale, float* __restrict__ shift)
{
    __shared__ float ssum[256];
    __shared__ float ssq[256];
    const int ch  = blockIdx.x;
    const int tid = threadIdx.x;
    const int total = Bq * Nq;   // 16384
    float s = 0.f, q = 0.f;
    for (int i = tid; i < total; i += 256) {
        const int bb = i >> 11;
        const int n  = i & (Nq - 1);
        const float v = data[((size_t)bb * CH + ch) * Nq + n];
        s += v; q += v * v;
    }
    ssum[tid] = s; ssq[tid] = q;
    __syncthreads();
    for (int off = 128; off > 0; off >>= 1) {
        if (tid < off) { ssum[tid] += ssum[tid + off]; ssq[tid] += ssq[tid + off]; }
        __syncthreads();
    }
    if (tid == 0) {
        const float inv  = 1.0f / (float)total;
        const float mean = ssum[0] * inv;
        const float var  = ssq[0] * inv - mean * mean;
        const float rstd = rsqrtf(var + 1e-5f);
        const float sc   = gamma[ch] * rstd;
        scale[ch] = sc;
        shift[ch] = beta[ch] - mean * sc;
    }
}

// ---------------- Kernel 3: BN1 + ReLU + conv1 (1x1, 96 -> 128) -----------------------
__global__ __launch_bounds__(256) void conv1_kernel(
    const float* __restrict__ out1, const float* __restrict__ w,
    const float* __restrict__ bias, const float* __restrict__ scale,
    const float* __restrict__ shift, float* __restrict__ out2)
{
    __shared__ float wl[INTERq * JCq];   // 48 KB
    const int tid = threadIdx.x;
    for (int i = tid; i < INTERq * JCq; i += 256) wl[i] = w[i];
    __syncthreads();

    const int p = blockIdx.x * 256 + tid;
    const int b = p >> 11;
    const int n = p & (Nq - 1);

    float z[JCq];
    #pragma unroll
    for (int jc = 0; jc < JCq; ++jc) {
        const float v = out1[((size_t)b * JCq + jc) * Nq + n];
        z[jc] = fmaxf(fmaf(scale[jc], v, shift[jc]), 0.f);
    }
    for (int i = 0; i < INTERq; ++i) {
        float acc = bias[i];
        #pragma unroll
        for (int jc = 0; jc < JCq; ++jc)
            acc = fmaf(wl[i * JCq + jc], z[jc], acc);
        out2[((size_t)b * INTERq + i) * Nq + n] = acc;
    }
}

// ---------------- Kernel 5: BN2 + ReLU + conv2 (128 -> 32) + concat with x ------------
__global__ __launch_bounds__(256) void conv2_concat_kernel(
    const float* __restrict__ out2, const float* __restrict__ w,
    const float* __restrict__ bias, const float* __restrict__ scale,
    const float* __restrict__ shift, const float* __restrict__ xg,
    float* __restrict__ outTail)
{
    __shared__ float wl[Fq * INTERq];   // 16 KB
    const int tid = threadIdx.x;
    for (int i = tid; i < Fq * INTERq; i += 256) wl[i] = w[i];
    __syncthreads();

    const int p = blockIdx.x * 256 + tid;
    const int b = p >> 11;
    const int n = p & (Nq - 1);

    float acc[Fq];
    #pragma unroll
    for (int f = 0; f < Fq; ++f) acc[f] = bias[f];

    for (int k = 0; k < INTERq; ++k) {
        float v = out2[((size_t)b * INTERq + k) * Nq + n];
        v = fmaxf(fmaf(scale[k], v, shift[k]), 0.f);
        #pragma unroll
        for (int f = 0; f < Fq; ++f)
            acc[f] = fmaf(wl[f * INTERq + k], v, acc[f]);
    }

    // concat: channels [0,32) = x, [32,64) = conv2 output
    #pragma unroll
    for (int c = 0; c < Cq; ++c)
        outTail[((size_t)b * (Cq + Fq) + c) * Nq + n] =
            xg[((size_t)b * Cq + c) * Nq + n];
    #pragma unroll
    for (int f = 0; f < Fq; ++f)
        outTail[((size_t)b * (Cq + Fq) + Cq + f) * Nq + n] = acc[f];
}

// --------------------------------- launcher -------------------------------------------
extern "C" void kernel_launch(void* const* d_in, const int* in_sizes, int n_in,
                              void* d_out, int out_size, void* d_ws, size_t ws_size,
                              hipStream_t stream) {
    const float* Wg  = (const float*)d_in[0];
    const float* xg  = (const float*)d_in[1];
    const float* g1  = (const float*)d_in[2];
    const float* be1 = (const float*)d_in[3];
    const float* w1  = (const float*)d_in[4];
    const float* b1  = (const float*)d_in[5];
    const float* g2  = (const float*)d_in[6];
    const float* be2 = (const float*)d_in[7];
    const float* w2  = (const float*)d_in[8];
    const float* b2  = (const float*)d_in[9];
    float* out = (float*)d_out;

    float* ws   = (float*)d_ws;
    float* out1 = ws;                                 // B*96*N  floats (6 MB)
    float* out2 = out1 + (size_t)Bq * JCq * Nq;       // B*128*N floats (8 MB)
    float* sc1  = out2 + (size_t)Bq * INTERq * Nq;
    float* sh1  = sc1 + JCq;
    float* sc2  = sh1 + JCq;
    float* sh2  = sc2 + INTERq;

    float* Wcopy = out;                               // first B*J*N*N elements
    float* tail  = out + (size_t)Bq * Jq * Nq * Nq;   // [B, C+F, N]

    gmul_wmma_kernel<<<Bq * Jq * (Nq / ROWS), 128, 0, stream>>>(Wg, xg, Wcopy, out1);
    bn_stats_kernel<<<JCq, 256, 0, stream>>>(out1, g1, be1, JCq, sc1, sh1);
    conv1_kernel<<<(Bq * Nq) / 256, 256, 0, stream>>>(out1, w1, b1, sc1, sh1, out2);
    bn_stats_kernel<<<INTERq, 256, 0, stream>>>(out2, g2, be2, INTERq, sc2, sh2);
    conv2_concat_kernel<<<(Bq * Nq) / 256, 256, 0, stream>>>(out2, w2, b2, sc2, sh2, xg, tail);
}